// GraphConvolution_15539191677217
// MI455X (gfx1250) — compile-verified
//
#include <hip/hip_runtime.h>

#define N_NODES 100000
#define FIN 512
#define FOUT 128

typedef __attribute__((ext_vector_type(2))) float v2f;
typedef __attribute__((ext_vector_type(8))) float v8f;

// ---------------------------------------------------------------------------
// Kernel 1: y = x @ theta  using V_WMMA_F32_16X16X4_F32 (full f32 precision).
// One wave computes one 16x16 output tile. Block = 256 threads = 8 waves,
// wave w handles N-tile w (8 * 16 = 128 = FOUT). grid.x = 100000/16 = 6250.
// ---------------------------------------------------------------------------
__global__ __launch_bounds__(256) void gemm_xtheta_wmma(
    const float* __restrict__ x,      // [N_NODES, FIN]
    const float* __restrict__ theta,  // [FIN, FOUT]
    float* __restrict__ y) {          // [N_NODES, FOUT]
  const int lane  = threadIdx.x & 31;
  const int wave  = threadIdx.x >> 5;          // N tile index 0..7
  const int mtile = blockIdx.x;                // 0..6249
  const int mlane = lane & 15;
  const int khalf = (lane >> 4) * 2;           // K offset: 0 (lanes 0-15) or 2 (lanes 16-31)
  const int ncol  = wave * 16 + mlane;

  const float* xrow = x + (size_t)(mtile * 16 + mlane) * FIN;

  v8f acc = {0.f, 0.f, 0.f, 0.f, 0.f, 0.f, 0.f, 0.f};

#pragma unroll 8
  for (int k = 0; k < FIN; k += 4) {
    // A 16x4: lane holds x[M=mlane(+tile)][k+khalf .. k+khalf+1] -> aligned float2
    v2f a = *(const v2f*)(xrow + k + khalf);
    // B 4x16: lane holds theta[k+khalf+0][ncol], theta[k+khalf+1][ncol]
    v2f b;
    b.x = theta[(size_t)(k + khalf + 0) * FOUT + ncol];
    b.y = theta[(size_t)(k + khalf + 1) * FOUT + ncol];
    acc = __builtin_amdgcn_wmma_f32_16x16x4_f32(
        /*neg_a=*/false, a, /*neg_b=*/false, b,
        /*c_mod=*/(short)0, acc, /*reuse_a=*/false, /*reuse_b=*/false);
  }

  // C/D layout: VGPR r -> M = r (lanes 0-15) or r+8 (lanes 16-31), N = lane%16
  float* yout = y + (size_t)(mtile * 16 + (lane >> 4) * 8) * FOUT + ncol - wave * 16 + wave * 16;
#pragma unroll
  for (int r = 0; r < 8; ++r)
    yout[(size_t)r * FOUT] = acc[r];
}

// ---------------------------------------------------------------------------
// Kernel 0: zero the accumulator (d_out), vectorized x4.
// ---------------------------------------------------------------------------
__global__ __launch_bounds__(256) void zero_out(float4* __restrict__ out, int n4) {
  int i = blockIdx.x * blockDim.x + threadIdx.x;
  if (i < n4) out[i] = make_float4(0.f, 0.f, 0.f, 0.f);
}

// ---------------------------------------------------------------------------
// Kernel 2: edge scatter  out[row] += val * y[col]   (128-wide rows).
// One wave per edge; lane l handles features [4l, 4l+3] via float4 load of y
// (y is 51.2 MB -> L2 resident) and 4 global_atomic_add_f32.
// ---------------------------------------------------------------------------
__global__ __launch_bounds__(256) void edge_scatter(
    const float* __restrict__ y,
    const int* __restrict__ rows,
    const int* __restrict__ cols,
    const float* __restrict__ vals,
    float* __restrict__ out,
    int n_edges) {
  int gid  = blockIdx.x * blockDim.x + threadIdx.x;
  int e    = gid >> 5;
  int lane = gid & 31;
  if (e >= n_edges) return;

  int   r = rows[e];
  int   c = cols[e];
  float v = vals[e];

  float4 yv = ((const float4*)(y + (size_t)c * FOUT))[lane];
  float* dst = out + (size_t)r * FOUT + lane * 4;
  atomicAdd(dst + 0, v * yv.x);
  atomicAdd(dst + 1, v * yv.y);
  atomicAdd(dst + 2, v * yv.z);
  atomicAdd(dst + 3, v * yv.w);
}

// ---------------------------------------------------------------------------
// Kernel 3: out = relu(out + bias), vectorized x4. (4*gid) % 128 -> bias4 idx.
// ---------------------------------------------------------------------------
__global__ __launch_bounds__(256) void bias_relu(float4* __restrict__ out,
                                                 const float4* __restrict__ bias4,
                                                 int n4) {
  int i = blockIdx.x * blockDim.x + threadIdx.x;
  if (i >= n4) return;
  float4 b = bias4[i & 31];  // (i*4) % 128 == (i%32)*4
  float4 o = out[i];
  o.x = fmaxf(o.x + b.x, 0.f);
  o.y = fmaxf(o.y + b.y, 0.f);
  o.z = fmaxf(o.z + b.z, 0.f);
  o.w = fmaxf(o.w + b.w, 0.f);
  out[i] = o;
}

// ---------------------------------------------------------------------------
extern "C" void kernel_launch(void* const* d_in, const int* in_sizes, int n_in,
                              void* d_out, int out_size, void* d_ws, size_t ws_size,
                              hipStream_t stream) {
  const float* x     = (const float*)d_in[0];  // [100000, 512]
  const int*   rows  = (const int*)  d_in[1];  // [E]
  const int*   cols  = (const int*)  d_in[2];  // [E]
  const float* vals  = (const float*)d_in[3];  // [E]
  const float* theta = (const float*)d_in[4];  // [512, 128]
  const float* bias  = (const float*)d_in[5];  // [128]
  float*       out   = (float*)d_out;          // [100000, 128]
  float*       y     = (float*)d_ws;           // scratch: [100000, 128] = 51.2 MB

  const int n_edges = in_sizes[1];
  const int n4      = (N_NODES * FOUT) / 4;    // 3.2M float4

  // 0) zero accumulator
  zero_out<<<(n4 + 255) / 256, 256, 0, stream>>>((float4*)out, n4);

  // 1) y = x @ theta  (WMMA f32)
  gemm_xtheta_wmma<<<N_NODES / 16, 256, 0, stream>>>(x, theta, y);

  // 2) out[row] += val * y[col]  (atomic scatter, L2-resident y and out)
  long long scatter_threads = (long long)n_edges * 32;
  int scatter_blocks = (int)((scatter_threads + 255) / 256);
  edge_scatter<<<scatter_blocks, 256, 0, stream>>>(y, rows, cols, vals, out, n_edges);

  // 3) out = relu(out + bias)
  bias_relu<<<(n4 + 255) / 256, 256, 0, stream>>>((float4*)out, (const float4*)bias, n4);
}